// CrispComposition_28183575396611
// MI455X (gfx1250) — compile-verified
//
#include <hip/hip_runtime.h>

// Problem dims (from reference)
#define BATCH 256
#define IN_F  1024
#define OUT_F 1024

// Tiling: 256-thread (8-wave32) blocks. Each block computes a 32x128 output
// tile over a 256-wide K chunk; 4 K-chunks are merged via float max atomics.
#define B_TILE   32
#define O_TILE   128
#define K_TILE   64
#define K_CHUNKS 4
#define K_PER_BLOCK (IN_F / K_CHUNKS)   // 256
#define N_TILES  (K_PER_BLOCK / K_TILE) // 4

// ---------------- CDNA5 type helpers ---------------------------------------
#if defined(__AMDGCN__)
typedef int v4i_t __attribute__((ext_vector_type(4)));
typedef __attribute__((address_space(1))) v4i_t gv4i_t;   // global int4
typedef __attribute__((address_space(3))) v4i_t lv4i_t;   // LDS int4
typedef unsigned int u32x4 __attribute__((ext_vector_type(4)));
typedef int i32x8 __attribute__((ext_vector_type(8)));
typedef int i32x4 __attribute__((ext_vector_type(4)));
#if __has_builtin(__builtin_amdgcn_tensor_load_to_lds) && __has_builtin(__builtin_amdgcn_s_wait_tensorcnt)
#define HAVE_TDM 1
#endif
#endif

// ---------------- async global->LDS (per-lane, ASYNCcnt) -------------------
__device__ __forceinline__ void async_copy_b128(const float* gsrc, float* ldst) {
#if defined(__AMDGCN__)
#if __has_builtin(__builtin_amdgcn_global_load_async_to_lds_b128)
  __builtin_amdgcn_global_load_async_to_lds_b128(
      (gv4i_t*)(v4i_t*)(void*)gsrc,
      (lv4i_t*)(v4i_t*)(void*)ldst,
      /*offset=*/0, /*cpol=*/0);
#else
  unsigned lds_off = (unsigned)(unsigned long long)(void*)ldst;
  asm volatile("global_load_async_to_lds_b128 %0, %1, off"
               :: "v"(lds_off), "v"(gsrc) : "memory");
#endif
#else
  (void)gsrc; (void)ldst;
#endif
}

__device__ __forceinline__ void wait_async0() {
#if defined(__AMDGCN__)
#if __has_builtin(__builtin_amdgcn_s_wait_asynccnt)
  __builtin_amdgcn_s_wait_asynccnt(0);
#else
  asm volatile("s_wait_asynccnt 0" ::: "memory");
#endif
#endif
}

// ---------------- TDM: one tensor_load_to_lds per 2D tile (TENSORcnt) ------
#if defined(HAVE_TDM)
// D# descriptor per CDNA5 ISA: group0 = control/lds/global
// addr, group1 = dims/tile/strides, groups 2-3 zero (2D tensor).
// This toolchain exposes the 6-arg builtin:
//   (uint32x4 g0, int32x8 g1, int32x4, int32x4, int32x8, i32 cpol)
__device__ __forceinline__ void tdm_load_tile_2d(const float* tile_base,
                                                 unsigned lds_off,
                                                 unsigned tile_cols,
                                                 unsigned tile_rows,
                                                 unsigned tensor_d0,
                                                 unsigned tensor_d1,
                                                 unsigned row_stride_elems) {
  const unsigned long long ga = (unsigned long long)(const void*)tile_base;
  u32x4 g0;
  g0.s0 = 1u;                                                 // count=1 (valid user D#)
  g0.s1 = lds_off;                                            // lds_addr (bytes)
  g0.s2 = (unsigned)ga;                                       // global_addr[31:0]
  g0.s3 = (unsigned)((ga >> 32) & 0x01FFFFFFu) | (2u << 30);  // addr[56:32] | type=2

  i32x8 g1;
  g1.s0 = (int)(2u << 16);                                    // data_size=2 (4B)
  g1.s1 = (int)((tensor_d0 & 0xFFFFu) << 16);                 // dim0 lo16 (bar addr=0)
  g1.s2 = (int)(((tensor_d0 >> 16) & 0xFFFFu) |
                ((tensor_d1 & 0xFFFFu) << 16));               // dim0 hi16 | dim1 lo16
  g1.s3 = (int)(((tensor_d1 >> 16) & 0xFFFFu) |
                (tile_cols << 16));                           // dim1 hi16 | tile_dim0
  g1.s4 = (int)(tile_rows & 0xFFFFu);                         // tile_dim1; tile_dim2=0
  g1.s5 = (int)row_stride_elems;                              // dim0_stride lo32
  g1.s6 = 0;                                                  // dim0_stride hi16 | dim1_stride lo16
  g1.s7 = 0;                                                  // dim1_stride hi32
  const i32x4 z4 = {0, 0, 0, 0};
  const i32x8 z8 = {0, 0, 0, 0, 0, 0, 0, 0};
  __builtin_amdgcn_tensor_load_to_lds(g0, g1, z4, z4, z8, 0);
}
#endif

// ---------------- Kernels ---------------------------------------------------
__global__ __launch_bounds__(256) void crisp_init_out(float* __restrict__ out) {
  const int idx = (blockIdx.x * 256 + threadIdx.x) * 4;
  const float ninf = -__builtin_inff();
  float4 v = make_float4(ninf, ninf, ninf, ninf);
  *reinterpret_cast<float4*>(&out[idx]) = v;
}

__global__ __launch_bounds__(256)
void crisp_maxmin_partial(const float* __restrict__ m,
                          const float* __restrict__ w,
                          float* __restrict__ out) {
  __shared__ float wT[2][K_TILE][O_TILE];   // 64 KB, double buffered
  __shared__ float mT[2][B_TILE][K_TILE];   // 16 KB, double buffered

  const int tid   = threadIdx.x;
  const int tx    = tid & 31;   // 32 col groups x 4 cols = 128
  const int ty    = tid >> 5;   // 8 row groups x 4 rows = 32
  const int o0    = blockIdx.x * O_TILE;
  const int b0    = blockIdx.y * B_TILE;
  const int kbase = blockIdx.z * K_PER_BLOCK;
  const bool wave0 = (ty == 0);  // wave-uniform gate for TDM issue

  float acc[4][4];
#pragma unroll
  for (int j = 0; j < 4; ++j)
#pragma unroll
    for (int c = 0; c < 4; ++c) acc[j][c] = -__builtin_inff();

  // Stage tile `it` into buffer `buf` (W via TDM DMA, m via async-to-LDS).
  auto stage = [&](int it, int buf) {
    const int i0 = kbase + it * K_TILE;
#if defined(HAVE_TDM)
    if (wave0) {
      tdm_load_tile_2d(&w[i0 * OUT_F + o0],
                       (unsigned)(unsigned long long)(void*)&wT[buf][0][0],
                       /*tile_cols=*/O_TILE, /*tile_rows=*/K_TILE,
                       /*tensor_d0=*/OUT_F, /*tensor_d1=*/IN_F,
                       /*row_stride=*/OUT_F);
    }
#else
#pragma unroll
    for (int t = 0; t < 8; ++t) {
      const int q = tid + t * 256;       // 0..2047 chunks of 16B
      const int r = q >> 5;              // 0..63
      const int c = (q & 31) << 2;       // 0..124 step 4
      async_copy_b128(&w[(i0 + r) * OUT_F + o0 + c], &wT[buf][r][c]);
    }
#endif
#pragma unroll
    for (int t = 0; t < 2; ++t) {
      const int q = tid + t * 256;       // 0..511 chunks of 16B
      const int b = q >> 4;              // 0..31
      const int k = (q & 15) << 2;       // 0..60 step 4
      async_copy_b128(&m[(b0 + b) * IN_F + i0 + k], &mT[buf][b][k]);
    }
  };

  stage(0, 0);

  for (int it = 0; it < N_TILES; ++it) {
    const int buf = it & 1;

    // Complete this tile's DMA before anyone reads it.
#if defined(HAVE_TDM)
    if (wave0) __builtin_amdgcn_s_wait_tensorcnt(0);
#endif
    wait_async0();
    __syncthreads();   // tile `it` visible to all waves; tile it-1 compute done

    // Kick off next tile's DMA into the other buffer; overlaps with compute.
    if (it + 1 < N_TILES) stage(it + 1, buf ^ 1);

    // 32 VALU ops (16 min + 16 max) per k-step per thread vs 3 DS loads.
#pragma unroll 2
    for (int k = 0; k < K_TILE; ++k) {
      const float4 wv = *reinterpret_cast<const float4*>(&wT[buf][k][tx << 2]);
#pragma unroll
      for (int j = 0; j < 4; ++j) {
        const float mv = mT[buf][(ty << 2) + j][k];   // broadcast across lanes
        acc[j][0] = fmaxf(acc[j][0], fminf(mv, wv.x));
        acc[j][1] = fmaxf(acc[j][1], fminf(mv, wv.y));
        acc[j][2] = fmaxf(acc[j][2], fminf(mv, wv.z));
        acc[j][3] = fmaxf(acc[j][3], fminf(mv, wv.w));
      }
    }
  }

  // Merge this K-chunk's partial maxima: native f32 max atomics (L2-resident).
#pragma unroll
  for (int j = 0; j < 4; ++j) {
    float* row = &out[(b0 + (ty << 2) + j) * OUT_F + o0 + (tx << 2)];
#pragma unroll
    for (int c = 0; c < 4; ++c) {
      __hip_atomic_fetch_max(row + c, acc[j][c],
                             __ATOMIC_RELAXED, __HIP_MEMORY_SCOPE_AGENT);
    }
  }
}

// ---------------- Host entry ------------------------------------------------
extern "C" void kernel_launch(void* const* d_in, const int* in_sizes, int n_in,
                              void* d_out, int out_size, void* d_ws, size_t ws_size,
                              hipStream_t stream) {
  (void)in_sizes; (void)n_in; (void)out_size; (void)d_ws; (void)ws_size;
  const float* m   = (const float*)d_in[0];   // [BATCH, IN_F]  crisp 0/1
  const float* w   = (const float*)d_in[1];   // [IN_F, OUT_F]
  float*       out = (float*)d_out;           // [BATCH, OUT_F]

  // Re-initialize output to -inf every call (deterministic, replay-safe).
  crisp_init_out<<<dim3((BATCH * OUT_F) / (256 * 4)), dim3(256), 0, stream>>>(out);

  dim3 grid(OUT_F / O_TILE, BATCH / B_TILE, K_CHUNKS);  // 8 x 8 x 4 = 256 blocks
  crisp_maxmin_partial<<<grid, dim3(256), 0, stream>>>(m, w, out);
}